// PagedLinear_11081015624230
// MI455X (gfx1250) — compile-verified
//
#include <hip/hip_runtime.h>

// out = x @ W^T + bias
// x: [B=4, S=2048, D_IN=4096] f32 -> M = 8192 rows
// W: [D_OUT=16384, D_IN=4096] f32, bias: [D_OUT] f32
// out: [M, D_OUT] f32

#define D_IN   4096u
#define D_OUT  16384u
#define M_TOT  8192u

typedef float v2f __attribute__((ext_vector_type(2)));
typedef float v8f __attribute__((ext_vector_type(8)));

static __device__ __forceinline__ v8f wmma4(v2f a, v2f b, v8f c) {
    return __builtin_amdgcn_wmma_f32_16x16x4_f32(false, a, false, b,
                                                 (short)0, c, false, false);
}

// Wave tile: 32(M) x 64(N): 8 accumulators, 8 WMMA per 6 fragment loads.
// Block: 256 threads = 8 waves, 2(M) x 4(N) -> block tile 64 x 256.
//
// K (=kd) is a RUNTIME argument so the compiler cannot fully unroll the K
// loop (keeps the steady-state body I-cache resident). The inner loop is an
// explicit PING-PONG over two fragment buffers (two K-steps per iteration):
// no loop-carried register rotation copies, and each WMMA group overlaps the
// in-flight loads of the other buffer.
__global__ __launch_bounds__(256)
void paged_linear_wmma_f32(const float* __restrict__ x,
                           const float* __restrict__ W,
                           const float* __restrict__ bias,
                           float* __restrict__ out,
                           int kd)
{
    const unsigned lane = threadIdx.x & 31u;
    const unsigned wave = threadIdx.x >> 5;
    const unsigned wm   = wave & 1u;    // 0..1 : M sub-tile
    const unsigned wn   = wave >> 1;    // 0..3 : N sub-tile

    const unsigned m0 = blockIdx.y * 64u  + wm * 32u;
    const unsigned n0 = blockIdx.x * 256u + wn * 64u;

    const unsigned l15   = lane & 15u;
    const unsigned khalf = (lane >> 4) * 2u;  // lanes 0-15: K,K+1 ; lanes 16-31: K+2,K+3

    // Per-lane base pointers; row deltas (16*D_IN*4 = 256KB) and K-step deltas
    // fold into the 24-bit instruction offset field.
    const float* pA = x + (size_t)(m0 + l15) * D_IN + khalf;
    const float* pB = W + (size_t)(n0 + l15) * D_IN + khalf;

    v8f acc[2][4];
    #pragma unroll
    for (int i = 0; i < 2; ++i)
        #pragma unroll
        for (int j = 0; j < 4; ++j)
            acc[i][j] = (v8f){};

    // Ping-pong fragment buffers (constant indices -> fixed registers).
    v2f a0[2], a1[2], b0[2], b1[2], b2[2], b3[2];

    // ---- prologue: buffer 0 <- fragments for k = 0 ----
    a0[0] = *(const v2f*)(pA);
    a1[0] = *(const v2f*)(pA + 16u * D_IN);
    b0[0] = *(const v2f*)(pB);
    b1[0] = *(const v2f*)(pB + 16u * D_IN);
    b2[0] = *(const v2f*)(pB + 32u * D_IN);
    b3[0] = *(const v2f*)(pB + 48u * D_IN);

    // ---- main loop: two K-steps per iteration ----
    for (int k = 0; k < kd - 8; k += 8) {
        // buffer 1 <- fragments for k+4
        a0[1] = *(const v2f*)(pA + 4u);
        a1[1] = *(const v2f*)(pA + 4u + 16u * D_IN);
        b0[1] = *(const v2f*)(pB + 4u);
        b1[1] = *(const v2f*)(pB + 4u + 16u * D_IN);
        b2[1] = *(const v2f*)(pB + 4u + 32u * D_IN);
        b3[1] = *(const v2f*)(pB + 4u + 48u * D_IN);

        // WMMA on buffer 0 (k)
        acc[0][0] = wmma4(a0[0], b0[0], acc[0][0]);
        acc[0][1] = wmma4(a0[0], b1[0], acc[0][1]);
        acc[0][2] = wmma4(a0[0], b2[0], acc[0][2]);
        acc[0][3] = wmma4(a0[0], b3[0], acc[0][3]);
        acc[1][0] = wmma4(a1[0], b0[0], acc[1][0]);
        acc[1][1] = wmma4(a1[0], b1[0], acc[1][1]);
        acc[1][2] = wmma4(a1[0], b2[0], acc[1][2]);
        acc[1][3] = wmma4(a1[0], b3[0], acc[1][3]);

        // buffer 0 <- fragments for k+8
        a0[0] = *(const v2f*)(pA + 8u);
        a1[0] = *(const v2f*)(pA + 8u + 16u * D_IN);
        b0[0] = *(const v2f*)(pB + 8u);
        b1[0] = *(const v2f*)(pB + 8u + 16u * D_IN);
        b2[0] = *(const v2f*)(pB + 8u + 32u * D_IN);
        b3[0] = *(const v2f*)(pB + 8u + 48u * D_IN);

        // WMMA on buffer 1 (k+4)
        acc[0][0] = wmma4(a0[1], b0[1], acc[0][0]);
        acc[0][1] = wmma4(a0[1], b1[1], acc[0][1]);
        acc[0][2] = wmma4(a0[1], b2[1], acc[0][2]);
        acc[0][3] = wmma4(a0[1], b3[1], acc[0][3]);
        acc[1][0] = wmma4(a1[1], b0[1], acc[1][0]);
        acc[1][1] = wmma4(a1[1], b1[1], acc[1][1]);
        acc[1][2] = wmma4(a1[1], b2[1], acc[1][2]);
        acc[1][3] = wmma4(a1[1], b3[1], acc[1][3]);

        pA += 8; pB += 8;
    }

    // ---- tail: buffer 0 holds k = kd-8; load kd-4 into buffer 1 ----
    a0[1] = *(const v2f*)(pA + 4u);
    a1[1] = *(const v2f*)(pA + 4u + 16u * D_IN);
    b0[1] = *(const v2f*)(pB + 4u);
    b1[1] = *(const v2f*)(pB + 4u + 16u * D_IN);
    b2[1] = *(const v2f*)(pB + 4u + 32u * D_IN);
    b3[1] = *(const v2f*)(pB + 4u + 48u * D_IN);

    acc[0][0] = wmma4(a0[0], b0[0], acc[0][0]);
    acc[0][1] = wmma4(a0[0], b1[0], acc[0][1]);
    acc[0][2] = wmma4(a0[0], b2[0], acc[0][2]);
    acc[0][3] = wmma4(a0[0], b3[0], acc[0][3]);
    acc[1][0] = wmma4(a1[0], b0[0], acc[1][0]);
    acc[1][1] = wmma4(a1[0], b1[0], acc[1][1]);
    acc[1][2] = wmma4(a1[0], b2[0], acc[1][2]);
    acc[1][3] = wmma4(a1[0], b3[0], acc[1][3]);

    acc[0][0] = wmma4(a0[1], b0[1], acc[0][0]);
    acc[0][1] = wmma4(a0[1], b1[1], acc[0][1]);
    acc[0][2] = wmma4(a0[1], b2[1], acc[0][2]);
    acc[0][3] = wmma4(a0[1], b3[1], acc[0][3]);
    acc[1][0] = wmma4(a1[1], b0[1], acc[1][0]);
    acc[1][1] = wmma4(a1[1], b1[1], acc[1][1]);
    acc[1][2] = wmma4(a1[1], b2[1], acc[1][2]);
    acc[1][3] = wmma4(a1[1], b3[1], acc[1][3]);

    // ---- epilogue: bias add + store ----
    // C/D 16x16 f32 layout: VGPR r -> M = r (lanes 0-15) / M = 8+r (lanes 16-31);
    // N = lane & 15.
    const unsigned rbase = (lane >> 4) * 8u;

    #pragma unroll
    for (int j = 0; j < 4; ++j) {
        const unsigned col = n0 + 16u * (unsigned)j + l15;
        const float bv = bias[col];
        // one base pointer per N-subtile; row strides become immediate offsets
        float* po = out + (size_t)(m0 + rbase) * D_OUT + col;
        #pragma unroll
        for (int r = 0; r < 8; ++r) {
            po[(unsigned)r * D_OUT]               = acc[0][j][r] + bv;  // rows m0+rbase+r
            po[(unsigned)r * D_OUT + 16u * D_OUT] = acc[1][j][r] + bv;  // rows +16
        }
    }
}

extern "C" void kernel_launch(void* const* d_in, const int* in_sizes, int n_in,
                              void* d_out, int out_size, void* d_ws, size_t ws_size,
                              hipStream_t stream) {
    const float* x    = (const float*)d_in[0];
    const float* W    = (const float*)d_in[1];
    const float* bias = (const float*)d_in[2];
    float* out        = (float*)d_out;

    // K passed at runtime so the device compiler cannot fully unroll the
    // K loop (keeps the inner loop I-cache resident). K = 4096 here.
    const int kd = (n_in >= 3 && in_sizes[2] > 0) ? (in_sizes[1] / in_sizes[2]) : (int)D_IN;

    dim3 grid(D_OUT / 256u, M_TOT / 64u);  // (64, 128)
    dim3 block(256);
    paged_linear_wmma_f32<<<grid, block, 0, stream>>>(x, W, bias, out, kd);
}